// OptimizedSimpleGCN_28441273434160
// MI455X (gfx1250) — compile-verified
//
#include <hip/hip_runtime.h>

// ---------------- problem constants ----------------
#define NN  100000      // nodes
#define NE  1600000     // edges
#define NG  512         // graphs
#define DIM 128         // in/hidden dim

typedef float v2f __attribute__((ext_vector_type(2)));
typedef float v8f __attribute__((ext_vector_type(8)));

__device__ __forceinline__ void atomAddF(float* p, float v) {
  unsafeAtomicAdd(p, v);   // -> global_atomic_add_f32
}

// ---------------- small elementwise kernels ----------------
__global__ __launch_bounds__(256) void k_init_deg(float* __restrict__ deg) {
  int i = blockIdx.x * 256 + threadIdx.x;
  if (i < NN) deg[i] = 1.0f;            // self-loop contribution
}

__global__ __launch_bounds__(256) void k_degree(const int* __restrict__ dst,
                                                float* __restrict__ deg) {
  int e = blockIdx.x * 256 + threadIdx.x;
  if (e < NE) atomAddF(&deg[dst[e]], 1.0f);
}

__global__ __launch_bounds__(256) void k_dis(const float* __restrict__ deg,
                                             float* __restrict__ dis) {
  int i = blockIdx.x * 256 + threadIdx.x;
  if (i < NN) dis[i] = rsqrtf(fmaxf(deg[i], 1.0f));
}

__global__ __launch_bounds__(256) void k_zero(float* __restrict__ p, int n) {
  int i = blockIdx.x * 256 + threadIdx.x;
  if (i < n) p[i] = 0.0f;
}

// ---------------- GEMM: H[N,128] = X[N,128] @ W[128,128] (f32 WMMA) -------
// One wave computes a 16-row tile across all 128 output columns.
// A frag layout (16x4 f32): lane = {half=lane>>4, r=lane&15}; per lane the two
// A elements are K = kb, kb+1 with kb = 4*kstep + 2*half  -> contiguous 8B load.
// B frag (4x16): element (K = kb+v, N = 16*n + r).  W is staged into LDS in a
// k-pair-interleaved layout so each B fragment is ONE ds_load_b64:
//   WsPair[((k>>1)*DIM + col)*2 + (k&1)] = W[k][col]
__global__ __launch_bounds__(128)
void k_gemm_wmma(const float* __restrict__ X, const float* __restrict__ W,
                 float* __restrict__ H, int nrows) {
  __shared__ float Ws[DIM * DIM];   // 64 KB: whole weight matrix, pair-interleaved

  const int tid = threadIdx.x;
  // cooperative stage of W into LDS with k-pair interleaving
  for (int idx = tid; idx < DIM * DIM; idx += 128) {
    int k   = idx >> 7;        // row of W (K index)
    int col = idx & (DIM - 1); // output column
    Ws[(((k >> 1) * DIM + col) << 1) + (k & 1)] = W[idx];
  }
  __syncthreads();

  const int wave = tid >> 5;
  const int lane = tid & 31;
  const int r    = lane & 15;     // row-in-tile for A, col-in-tile for B
  const int h    = lane >> 4;     // lane half
  const int row0 = blockIdx.x * 64 + wave * 16;

  if (row0 < nrows) {             // N is a multiple of 16 -> tile all-or-nothing
    v8f acc[8] = {};              // 8 x (16x16 f32) accumulators = full 128 cols
    const float* xrow = X + (size_t)(row0 + r) * DIM;

    for (int ks = 0; ks < 32; ++ks) {        // K = 128 in steps of 4
      const int kb = ks * 4 + 2 * h;         // even by construction
      v2f a = *(const v2f*)(xrow + kb);      // contiguous 8B global load
      const float* bbase = Ws + (((kb >> 1) * DIM + r) << 1);
#pragma unroll
      for (int n = 0; n < 8; ++n) {
        v2f b = *(const v2f*)(bbase + (n << 5));   // one ds_load_b64
        acc[n] = __builtin_amdgcn_wmma_f32_16x16x4_f32(
            false, a, false, b, (short)0, acc[n], false, false);
      }
    }

    // C/D layout: VGPR v, lanes0-15 -> M=v, lanes16-31 -> M=v+8; N=r
#pragma unroll
    for (int n = 0; n < 8; ++n) {
#pragma unroll
      for (int v = 0; v < 8; ++v) {
        const int m = v + 8 * h;
        H[(size_t)(row0 + m) * DIM + n * 16 + r] = acc[n][v];
      }
    }
  }
}

// ---------------- aggregation ----------------
// agg[i] = h[i] * dis[i]^2   (self-loop term; also fully initializes agg)
__global__ __launch_bounds__(256)
void k_self_init(const float* __restrict__ h, const float* __restrict__ dis,
                 float* __restrict__ agg) {
  int t = blockIdx.x * 256 + threadIdx.x;   // NN*32 threads, float4 each
  int node = t >> 5, lane = t & 31;
  if (node < NN) {
    float s = dis[node]; s *= s;
    float4 v = ((const float4*)(h + (size_t)node * DIM))[lane];
    v.x *= s; v.y *= s; v.z *= s; v.w *= s;
    ((float4*)(agg + (size_t)node * DIM))[lane] = v;
  }
}

// one wave per edge: agg[dst] += h[src] * dis[src]*dis[dst]
__global__ __launch_bounds__(256)
void k_edge_agg(const int* __restrict__ src, const int* __restrict__ dst,
                const float* __restrict__ dis, const float* __restrict__ h,
                float* __restrict__ agg) {
  int t = blockIdx.x * 256 + threadIdx.x;
  int e = t >> 5, lane = t & 31;
  if (e < NE) {
    int s = src[e], d = dst[e];
    float norm = dis[s] * dis[d];
    float4 v = ((const float4*)(h + (size_t)s * DIM))[lane];
    float* out = agg + (size_t)d * DIM + lane * 4;
    atomAddF(out + 0, v.x * norm);
    atomAddF(out + 1, v.y * norm);
    atomAddF(out + 2, v.z * norm);
    atomAddF(out + 3, v.w * norm);
  }
}

__global__ __launch_bounds__(256)
void k_bias_relu(float* __restrict__ agg, const float* __restrict__ b) {
  int t = blockIdx.x * 256 + threadIdx.x;   // NN*DIM threads
  if (t < NN * DIM) {
    int f = t & (DIM - 1);
    agg[t] = fmaxf(agg[t] + b[f], 0.0f);
  }
}

// ---------------- pooling + readout ----------------
__global__ __launch_bounds__(256)
void k_pool_sum(const float* __restrict__ h, const int* __restrict__ batch,
                float* __restrict__ sums) {
  int t = blockIdx.x * 256 + threadIdx.x;   // NN*32 threads, float4 each
  int node = t >> 5, lane = t & 31;
  if (node < NN) {
    int g = batch[node];
    float4 v = ((const float4*)(h + (size_t)node * DIM))[lane];
    float* out = sums + (size_t)g * DIM + lane * 4;
    atomAddF(out + 0, v.x);
    atomAddF(out + 1, v.y);
    atomAddF(out + 2, v.z);
    atomAddF(out + 3, v.w);
  }
}

__global__ __launch_bounds__(256)
void k_count(const int* __restrict__ batch, float* __restrict__ cnt) {
  int i = blockIdx.x * 256 + threadIdx.x;
  if (i < NN) atomAddF(&cnt[batch[i]], 1.0f);
}

__global__ __launch_bounds__(256)
void k_final(const float* __restrict__ sums, const float* __restrict__ cnt,
             const float* __restrict__ Wp, const float* __restrict__ bp,
             float* __restrict__ out) {
  int g = blockIdx.x * 256 + threadIdx.x;
  if (g < NG) {
    float inv = 1.0f / fmaxf(cnt[g], 1.0f);
    float acc = bp[0];
    const float* s = sums + (size_t)g * DIM;
    for (int f = 0; f < DIM; ++f) acc += s[f] * inv * Wp[f];
    out[g] = acc;
  }
}

// ---------------- host orchestration ----------------
extern "C" void kernel_launch(void* const* d_in, const int* in_sizes, int n_in,
                              void* d_out, int out_size, void* d_ws, size_t ws_size,
                              hipStream_t stream) {
  const float* x    = (const float*)d_in[0];
  const int*   ei   = (const int*)d_in[1];   // [2, NE] flat
  const int*   batch= (const int*)d_in[2];
  const float* W1   = (const float*)d_in[3];
  const float* b1   = (const float*)d_in[4];
  const float* W2   = (const float*)d_in[5];
  const float* b2   = (const float*)d_in[6];
  const float* Wp   = (const float*)d_in[7];
  const float* bp   = (const float*)d_in[8];
  float*       out  = (float*)d_out;

  const int* srcp = ei;
  const int* dstp = ei + NE;

  // workspace layout
  const size_t FEAT = (size_t)NN * DIM;         // 12.8M floats
  char* ws   = (char*)d_ws;
  float* hbuf = (float*)ws;                     // GEMM output
  float* abuf = (float*)(ws + FEAT * 4);        // aggregation buffer
  float* deg  = (float*)(ws + 2 * FEAT * 4);    // NN
  float* dis  = deg + NN;                       // NN
  float* sums = dis + NN;                       // NG*DIM
  float* cnts = sums + (size_t)NG * DIM;        // NG  (contiguous after sums)

  const int B = 256;
  dim3 blk(B);
  int gNodes   = (NN + B - 1) / B;
  int gEdges   = (NE + B - 1) / B;
  int gNode32  = (NN * 32 + B - 1) / B;     // wave-per-node float4 kernels
  int gEdge32  = (int)(((size_t)NE * 32 + B - 1) / B);
  int gFeat    = (NN * DIM + B - 1) / B;
  int gGemm    = (NN + 63) / 64;            // 64 rows per 128-thread block

  // degrees (with self loops) and normalization
  k_init_deg<<<gNodes, blk, 0, stream>>>(deg);
  k_degree  <<<gEdges, blk, 0, stream>>>(dstp, deg);
  k_dis     <<<gNodes, blk, 0, stream>>>(deg, dis);

  // ---- layer 1 ----
  k_gemm_wmma<<<gGemm, dim3(128), 0, stream>>>(x, W1, hbuf, NN);
  k_self_init<<<gNode32, blk, 0, stream>>>(hbuf, dis, abuf);
  k_edge_agg <<<gEdge32, blk, 0, stream>>>(srcp, dstp, dis, hbuf, abuf);
  k_bias_relu<<<gFeat, blk, 0, stream>>>(abuf, b1);

  // ---- layer 2 ----
  k_gemm_wmma<<<gGemm, dim3(128), 0, stream>>>(abuf, W2, hbuf, NN);
  k_self_init<<<gNode32, blk, 0, stream>>>(hbuf, dis, abuf);
  k_edge_agg <<<gEdge32, blk, 0, stream>>>(srcp, dstp, dis, hbuf, abuf);
  k_bias_relu<<<gFeat, blk, 0, stream>>>(abuf, b2);

  // ---- mean pool + readout ----
  int nz = NG * DIM + NG;
  k_zero   <<<(nz + B - 1) / B, blk, 0, stream>>>(sums, nz);  // sums+cnts contiguous
  k_pool_sum<<<gNode32, blk, 0, stream>>>(abuf, batch, sums);
  k_count  <<<gNodes, blk, 0, stream>>>(batch, cnts);
  k_final  <<<(NG + B - 1) / B, blk, 0, stream>>>(sums, cnts, Wp, bp, out);
}